// Yolov1Loss_46119358824931
// MI455X (gfx1250) — compile-verified
//
#include <hip/hip_runtime.h>
#include <cstdint>
#include <cstddef>

// ---------------- problem constants (from reference) ----------------
constexpr int kB     = 16384;
constexpr int kS     = 7;
constexpr int kD     = 30;          // NB*5 + NC
constexpr int kNC    = 20;
constexpr int kCells = kB * kS * kS;        // 802816
constexpr int kTPB   = 256;
constexpr int kNBLK  = kCells / kTPB;       // 3136 (exact)

#define LDS_AS __attribute__((address_space(3)))

// -------- CDNA5 async global->LDS staging (ASYNCcnt path), via inline asm --------
// VFLAT async form: global_load_async_to_lds_b128 VDST(lds byte addr), VADDR(64b), off
static __device__ __forceinline__ void async_copy_b128(const float* g, float* l) {
  unsigned lds_addr = (unsigned)(size_t)(LDS_AS float*)l;   // 32-bit LDS offset
  asm volatile("global_load_async_to_lds_b128 %0, %1, off"
               :
               : "v"(lds_addr), "v"(g)
               : "memory");
}

static __device__ __forceinline__ void wait_async_zero() {
#if __has_builtin(__builtin_amdgcn_s_wait_asynccnt)
  __builtin_amdgcn_s_wait_asynccnt(0);
#else
  asm volatile("s_wait_asynccnt 0" ::: "memory");
#endif
}

static __device__ __forceinline__ float sigmoidf_(float x) {
  return 1.0f / (1.0f + __expf(-x));
}

// wave32 + cross-wave block reduction (deterministic, no atomics)
static __device__ __forceinline__ float block_reduce_sum(float v, float* sRed) {
  const int tid  = threadIdx.x;
  const int lane = tid & 31;
  const int wid  = tid >> 5;
  #pragma unroll
  for (int o = 16; o > 0; o >>= 1) v += __shfl_down(v, o, 32);
  if (lane == 0) sRed[wid] = v;
  __syncthreads();
  if (tid < 8) {
    v = sRed[tid];
    v += __shfl_down(v, 4, 32);
    v += __shfl_down(v, 2, 32);
    v += __shfl_down(v, 1, 32);
  }
  return v;  // valid in tid==0
}

__global__ __launch_bounds__(kTPB)
void yolo_loss_main(const float* __restrict__ pred,
                    const int*   __restrict__ grid,
                    const float* __restrict__ tbox,
                    const int*   __restrict__ tcls,
                    float*       __restrict__ partials) {
  __shared__ __align__(16) float sPred[kTPB * kD];   // 30 KB
  __shared__ float sRed[8];

  const int tid        = threadIdx.x;
  const int blockStart = blockIdx.x * kTPB;
  const int c          = blockStart + tid;          // this thread's cell

  // warm the scalar-ish side streams while async copies are in flight
  __builtin_prefetch(&tbox[(size_t)c * 4], 0, 3);
  __builtin_prefetch(&grid[c], 0, 3);
  __builtin_prefetch(&tcls[c], 0, 3);

  // stage this block's 256 cells * 30 floats = 1920 x 16B chunks into LDS.
  // base byte offset = blockIdx.x * 30720 -> 16B aligned.
  const float* gbase = pred + (size_t)blockStart * kD;
  for (int k = tid; k < (kTPB * kD) / 4; k += kTPB) {
    async_copy_b128(gbase + 4 * k, sPred + 4 * k);
  }
  wait_async_zero();
  __syncthreads();

  // ---------------- per-cell loss ----------------
  const int ij = c % (kS * kS);
  const float yg = (float)(ij / kS);   // row index (S0)
  const float xg = (float)(ij % kS);   // col index (S1)

  const float* P = &sPred[tid * kD];   // stride 30 floats: conflict-free banks

  const float conf0 = sigmoidf_(P[0]);
  const float conf1 = sigmoidf_(P[1]);

  float bx[2], by[2], bw[2], bh[2];
  #pragma unroll
  for (int nb = 0; nb < 2; ++nb) {
    bx[nb] = sigmoidf_(P[2 + 4 * nb + 0]);
    by[nb] = sigmoidf_(P[2 + 4 * nb + 1]);
    bw[nb] = sigmoidf_(P[2 + 4 * nb + 2]);
    bh[nb] = sigmoidf_(P[2 + 4 * nb + 3]);
  }

  // softmax: only the target-class probability is needed
  float m = P[10];
  #pragma unroll
  for (int q = 11; q < 10 + kNC; ++q) m = fmaxf(m, P[q]);
  float sum = 0.0f;
  #pragma unroll
  for (int q = 10; q < 10 + kNC; ++q) sum += __expf(P[q] - m);
  const int   t     = tcls[c];
  const float cls_t = __expf(P[10 + t] - m) / sum;

  const float4 tb  = ((const float4*)tbox)[c];
  const float tox = tb.x, toy = tb.y, tw = tb.z, th = tb.w;

  const float inv7 = 1.0f / 7.0f;
  const float tcx = (xg + tox) * inv7;
  const float tcy = (yg + toy) * inv7;
  const float tarea = tw * th;

  float iou[2];
  #pragma unroll
  for (int nb = 0; nb < 2; ++nb) {
    const float pcx = (xg + bx[nb]) * inv7;
    const float pcy = (yg + by[nb]) * inv7;
    const float pw = bw[nb], ph = bh[nb];
    const float tb_ = fminf(tcx + tw * 0.5f, pcx + pw * 0.5f) -
                      fmaxf(tcx - tw * 0.5f, pcx - pw * 0.5f);
    const float lr_ = fminf(tcy + th * 0.5f, pcy + ph * 0.5f) -
                      fmaxf(tcy - th * 0.5f, pcy - ph * 0.5f);
    const float inter = ((tb_ < 0.0f) || (lr_ < 0.0f)) ? 0.0f : tb_ * lr_;
    iou[nb] = inter / (tarea + pw * ph - inter);
  }

  const int best = (iou[1] > iou[0]) ? 1 : 0;   // argmax, first-on-ties
  const float dx = bx[best] - tox;
  const float dy = by[best] - toy;
  const float dw = sqrtf(bw[best]) - sqrtf(tw);
  const float dh = sqrtf(bh[best]) - sqrtf(th);
  const float coord = dx * dx + dy * dy + dw * dw + dh * dh;

  const float conf_b = best ? conf1 : conf0;
  const float dc = conf_b - iou[best];
  const float dk = 1.0f - cls_t;
  const float obj_loss   = 5.0f * coord + dc * dc + dk * dk;
  const float noobj_loss = 0.5f * (conf0 * conf0 + conf1 * conf1);

  const float v = (grid[c] == 1) ? obj_loss : noobj_loss;

  const float bsum = block_reduce_sum(v, sRed);
  if (tid == 0) partials[blockIdx.x] = bsum;
}

__global__ __launch_bounds__(kTPB)
void yolo_loss_reduce(const float* __restrict__ partials,
                      float*       __restrict__ out, int n) {
  __shared__ float sRed[8];
  float v = 0.0f;
  for (int k = threadIdx.x; k < n; k += kTPB) v += partials[k];
  const float total = block_reduce_sum(v, sRed);
  if (threadIdx.x == 0) out[0] = total * (1.0f / (float)kB);
}

extern "C" void kernel_launch(void* const* d_in, const int* in_sizes, int n_in,
                              void* d_out, int out_size, void* d_ws, size_t ws_size,
                              hipStream_t stream) {
  (void)in_sizes; (void)n_in; (void)out_size; (void)ws_size;
  const float* pred = (const float*)d_in[0];
  const int*   grid = (const int*)d_in[1];
  const float* tbox = (const float*)d_in[2];
  const int*   tcls = (const int*)d_in[3];
  float* out      = (float*)d_out;
  float* partials = (float*)d_ws;   // kNBLK floats (12.25 KB) of scratch

  yolo_loss_main<<<kNBLK, kTPB, 0, stream>>>(pred, grid, tbox, tcls, partials);
  yolo_loss_reduce<<<1, kTPB, 0, stream>>>(partials, out, kNBLK);
}